// StructureEmbed_3908420239568
// MI455X (gfx1250) — compile-verified
//
#include <hip/hip_runtime.h>
#include <stdint.h>

#define DIM 128
#define ROW4 (DIM / 4)          // 32 float4 per row == one wave32
#define WPB 8                   // waves per block
#define TPB (WPB * 32)
#define LROW (DIM + 4)          // LDS row stride (pad 4 floats: bank rotation)

typedef float v2f __attribute__((ext_vector_type(2)));
typedef float v8f __attribute__((ext_vector_type(8)));

#if defined(__has_builtin)
#  if __has_builtin(__builtin_amdgcn_wmma_f32_16x16x4_f32)
#    define HAVE_WMMA_F32X4 1
#  endif
#endif
#ifndef HAVE_WMMA_F32X4
#  define HAVE_WMMA_F32X4 0
#endif

// ---------------------------------------------------------------------------
// CDNA5 async edge streaming helpers (ASYNCcnt-tracked global->LDS DMA)
// ---------------------------------------------------------------------------
__device__ __forceinline__ void async_edge_load(const int2* gp, void* lds_ptr) {
  unsigned off = (unsigned)(uintptr_t)lds_ptr;           // low 32 bits = LDS offset
  unsigned long long ga = (unsigned long long)(uintptr_t)gp;
  asm volatile("global_load_async_to_lds_b64 %0, %1, off"
               :: "v"(off), "v"(ga) : "memory");
}
__device__ __forceinline__ void wait_async_le1() {
#if defined(__has_builtin) && __has_builtin(__builtin_amdgcn_s_wait_asynccnt)
  __builtin_amdgcn_s_wait_asynccnt(1);
#else
  asm volatile("s_wait_asynccnt 0x1" ::: "memory");
#endif
}
__device__ __forceinline__ void wait_async_le0() {
#if defined(__has_builtin) && __has_builtin(__builtin_amdgcn_s_wait_asynccnt)
  __builtin_amdgcn_s_wait_asynccnt(0);
#else
  asm volatile("s_wait_asynccnt 0x0" ::: "memory");
#endif
}

// ---------------------------------------------------------------------------
// Small utility kernels
// ---------------------------------------------------------------------------
__global__ void fill_f32(float* __restrict__ p, float v, int n) {
  int i = blockIdx.x * blockDim.x + threadIdx.x;
  int stride = gridDim.x * blockDim.x;
  for (; i < n; i += stride) p[i] = v;
}

// deg[e.x] += 1, deg[e.y] += 1  (deg pre-filled with 1.0 for the self loop)
__global__ void degree_accum(const int2* __restrict__ edges, int nE,
                             float* __restrict__ deg) {
  int i = blockIdx.x * blockDim.x + threadIdx.x;
  int stride = gridDim.x * blockDim.x;
  for (; i < nE; i += stride) {
    int2 e = edges[i];
    unsafeAtomicAdd(deg + e.x, 1.0f);
    unsafeAtomicAdd(deg + e.y, 1.0f);
  }
}

__global__ void rsqrt_inplace(float* __restrict__ p, int n) {
  int i = blockIdx.x * blockDim.x + threadIdx.x;
  int stride = gridDim.x * blockDim.x;
  for (; i < n; i += stride) p[i] = rsqrtf(p[i]);
}

// ---------------------------------------------------------------------------
// Edge scatter: one wave handles a 32-edge chunk; LDS double buffer fed by
// async DMA, feature rows prefetched, 32 lanes cover one 128-f32 row,
// both symmetric directions of each original edge handled together.
// agg[d] += x[s]*c ; agg[s] += x[d]*c  with c = dis[s]*dis[d]
// ---------------------------------------------------------------------------
__global__ __launch_bounds__(TPB) void gcn_scatter(
    const int2* __restrict__ edges, int nE,
    const float* __restrict__ dis,
    const float* __restrict__ x,
    float* __restrict__ agg) {
  __shared__ int2 ebuf[WPB][2][32];
  const int lane = threadIdx.x & 31;
  const int wv = threadIdx.x >> 5;
  const int nChunks = (nE + 31) >> 5;
  const int wgid = blockIdx.x * WPB + wv;
  const int wstride = gridDim.x * WPB;

  if (wgid < nChunks) {
    int g = wgid * 32 + lane; if (g >= nE) g = nE - 1;
    async_edge_load(edges + g, &ebuf[wv][0][lane]);
  }

  int buf = 0;
  for (int c = wgid; c < nChunks; c += wstride, buf ^= 1) {
    int nc = c + wstride;
    if (nc < nChunks) {                       // kick DMA for the next chunk
      int g = nc * 32 + lane; if (g >= nE) g = nE - 1;
      async_edge_load(edges + g, &ebuf[wv][buf ^ 1][lane]);
      wait_async_le1();                       // oldest (current chunk) landed
    } else {
      wait_async_le0();
    }

    int2 e = ebuf[wv][buf][lane];             // my edge of this chunk
    float cc = dis[e.x] * dis[e.y];           // per-edge norm coefficient

    // prefetch both feature rows of every edge in the chunk ahead of the
    // sequential inner loop (emits global_prefetch_b8)
    __builtin_prefetch(x + (size_t)e.x * DIM, 0, 2);
    __builtin_prefetch(x + (size_t)e.y * DIM, 0, 2);

    int m = nE - (c << 5); if (m > 32) m = 32;
    const int col = lane << 2;                // 4 floats per lane
    for (int j = 0; j < m; ++j) {
      int s   = __shfl(e.x, j, 32);
      int d   = __shfl(e.y, j, 32);
      float cj = __shfl(cc, j, 32);
      const float4 xs = *(const float4*)(x + (size_t)s * DIM + col);
      const float4 xd = *(const float4*)(x + (size_t)d * DIM + col);
      float* ad = agg + (size_t)d * DIM + col;
      float* as = agg + (size_t)s * DIM + col;
      unsafeAtomicAdd(ad + 0, xs.x * cj);
      unsafeAtomicAdd(ad + 1, xs.y * cj);
      unsafeAtomicAdd(ad + 2, xs.z * cj);
      unsafeAtomicAdd(ad + 3, xs.w * cj);
      unsafeAtomicAdd(as + 0, xd.x * cj);
      unsafeAtomicAdd(as + 1, xd.y * cj);
      unsafeAtomicAdd(as + 2, xd.z * cj);
      unsafeAtomicAdd(as + 3, xd.w * cj);
    }
  }
  wait_async_le0();
}

// x_out = relu(agg + x_in * dis[row]^2); also re-zero agg for the next layer
__global__ void combine_relu_zero(float* __restrict__ agg,
                                  const float* __restrict__ xin,
                                  const float* __restrict__ dis,
                                  float* __restrict__ xout, int n4) {
  int i = blockIdx.x * blockDim.x + threadIdx.x;
  int stride = gridDim.x * blockDim.x;
  const float4 z4 = make_float4(0.f, 0.f, 0.f, 0.f);
  for (; i < n4; i += stride) {
    int row = i >> 5;                         // ROW4 == 32
    float sc = dis[row]; sc *= sc;
    float4 a  = ((const float4*)agg)[i];
    float4 xv = ((const float4*)xin)[i];
    float4 v;
    v.x = fmaxf(fmaf(xv.x, sc, a.x), 0.f);
    v.y = fmaxf(fmaf(xv.y, sc, a.y), 0.f);
    v.z = fmaxf(fmaf(xv.z, sc, a.z), 0.f);
    v.w = fmaxf(fmaf(xv.w, sc, a.w), 0.f);
    ((float4*)xout)[i] = v;
    ((float4*)agg)[i]  = z4;                  // ready for layer 1
  }
}

// ---------------------------------------------------------------------------
// Final layer, fused: v = agg + x*dis^2, then row L2 norms for a 16-row tile
// via V_WMMA_F32_16X16X4_F32:  Gram = V*V^T accumulated over K=128, take diag.
// A(16x4) and B(4x16)=A^T have IDENTICAL per-lane register contents for the
// Gram product, so the same float2 feeds both operands. Tile is staged in
// padded LDS so all global traffic stays coalesced b128.
// ---------------------------------------------------------------------------
__global__ __launch_bounds__(TPB) void combine_norm_out(
    const float* __restrict__ agg, const float* __restrict__ xin,
    const float* __restrict__ dis, float* __restrict__ out, int nRows) {
  __shared__ float vbuf[WPB][16][LROW];
  const int lane = threadIdx.x & 31;
  const int wv = threadIdx.x >> 5;
  const int nTiles = (nRows + 15) >> 4;

  for (int tile = blockIdx.x * WPB + wv; tile < nTiles; tile += gridDim.x * WPB) {
    const int base = tile << 4;

    // stage v = agg + x*dis^2 for 16 rows into LDS (coalesced b128 reads)
#pragma unroll
    for (int r = 0; r < 16; ++r) {
      int row = base + r; if (row >= nRows) row = nRows - 1;
      float sc = dis[row]; sc *= sc;
      size_t i4 = (size_t)row * ROW4 + lane;
      float4 a  = ((const float4*)agg)[i4];
      float4 xv = ((const float4*)xin)[i4];
      float4 v;
      v.x = fmaf(xv.x, sc, a.x);
      v.y = fmaf(xv.y, sc, a.y);
      v.z = fmaf(xv.z, sc, a.z);
      v.w = fmaf(xv.w, sc, a.w);
      *(float4*)&vbuf[wv][r][lane << 2] = v;
    }

    float ssq[16];
#if HAVE_WMMA_F32X4
    {
      // A layout: lanes 0-15 = rows 0-15 with K=0,1; lanes 16-31 same rows K=2,3
      const int lrow = lane & 15;
      const int koff = (lane >> 4) << 1;      // 0 or 2
      v8f c = {};
#pragma unroll
      for (int t = 0; t < DIM / 4; ++t) {
        v2f a = *(const v2f*)&vbuf[wv][lrow][(t << 2) + koff];
        c = __builtin_amdgcn_wmma_f32_16x16x4_f32(
            /*neg_a=*/false, a, /*neg_b=*/false, a,
            /*c_mod=*/(short)0, c, /*reuse_a=*/false, /*reuse_b=*/false);
      }
      // diag(Gram): m<8 -> VGPR m, lane m ; m>=8 -> VGPR m-8, lane m+16
#pragma unroll
      for (int r = 0; r < 16; ++r)
        ssq[r] = (r < 8) ? __shfl(c[r], r, 32) : __shfl(c[r - 8], r + 16, 32);
    }
#else
    {
#pragma unroll
      for (int r = 0; r < 16; ++r) {
        float4 v = *(const float4*)&vbuf[wv][r][lane << 2];
        float ss = v.x * v.x + v.y * v.y + v.z * v.z + v.w * v.w;
#pragma unroll
        for (int o = 16; o > 0; o >>= 1) ss += __shfl_xor(ss, o, 32);
        ssq[r] = ss;
      }
    }
#endif

    // normalize and write out (coalesced b128 stores)
#pragma unroll
    for (int r = 0; r < 16; ++r) {
      int row = base + r;
      if (row < nRows) {
        float inv = 1.0f / fmaxf(sqrtf(ssq[r]), 1e-12f);
        float4 v = *(const float4*)&vbuf[wv][r][lane << 2];
        v.x *= inv; v.y *= inv; v.z *= inv; v.w *= inv;
        ((float4*)out)[(size_t)row * ROW4 + lane] = v;
      }
    }
  }
}

__global__ void gather_rows(const float* __restrict__ ent,
                            const int* __restrict__ seeds,
                            float* __restrict__ out, int nSeeds) {
  int i = blockIdx.x * blockDim.x + threadIdx.x;
  int stride = gridDim.x * blockDim.x;
  int n4 = nSeeds * ROW4;
  for (; i < n4; i += stride) {
    int row = i >> 5, k = i & 31;
    ((float4*)out)[i] = ((const float4*)ent)[(size_t)seeds[row] * ROW4 + k];
  }
}

// ---------------------------------------------------------------------------
extern "C" void kernel_launch(void* const* d_in, const int* in_sizes, int n_in,
                              void* d_out, int out_size, void* d_ws, size_t ws_size,
                              hipStream_t stream) {
  const int*   seeds[2] = {(const int*)d_in[0], (const int*)d_in[1]};
  const float* feats[2] = {(const float*)d_in[2], (const float*)d_in[3]};
  const int2*  edges[2] = {(const int2*)d_in[4], (const int2*)d_in[5]};
  const int nSeeds = in_sizes[0];
  const int N = in_sizes[2] / DIM;
  const int E = in_sizes[4] / 2;

  float* out = (float*)d_out;
  float* seed_out[2];
  float* ent_out[2];
  seed_out[0] = out;
  seed_out[1] = out + (size_t)nSeeds * DIM;
  ent_out[0]  = out + (size_t)2 * nSeeds * DIM;
  ent_out[1]  = ent_out[0] + (size_t)N * DIM;

  // workspace: dis[N] | x1[N*DIM] | agg[N*DIM]   (graphs processed serially)
  float* dis = (float*)d_ws;
  float* x1  = dis + N;
  float* agg = x1 + (size_t)N * DIM;

  const int nd  = N * DIM;
  const int n4  = N * ROW4;
  const int nTiles = (N + 15) / 16;
  const int blkFill  = (nd + TPB - 1) / TPB;
  const int blkN     = (N + TPB - 1) / TPB;
  const int blkE     = (E + TPB - 1) / TPB;
  const int blkComb  = (n4 + TPB - 1) / TPB;
  const int blkNorm  = (nTiles + WPB - 1) / WPB;
  const int blkGath  = (nSeeds * ROW4 + TPB - 1) / TPB;
  const int blkScat  = 1280;   // 10240 waves over ~25000 chunks

  for (int g = 0; g < 2; ++g) {
    // degree (dis doubles as deg), agg = 0
    fill_f32<<<blkN, TPB, 0, stream>>>(dis, 1.0f, N);
    fill_f32<<<blkFill, TPB, 0, stream>>>(agg, 0.0f, nd);
    degree_accum<<<blkE, TPB, 0, stream>>>(edges[g], E, dis);
    rsqrt_inplace<<<blkN, TPB, 0, stream>>>(dis, N);
    // layer 0
    gcn_scatter<<<blkScat, TPB, 0, stream>>>(edges[g], E, dis, feats[g], agg);
    combine_relu_zero<<<blkComb, TPB, 0, stream>>>(agg, feats[g], dis, x1, n4);
    // layer 1
    gcn_scatter<<<blkScat, TPB, 0, stream>>>(edges[g], E, dis, x1, agg);
    combine_norm_out<<<blkNorm, TPB, 0, stream>>>(agg, x1, dis, ent_out[g], N);
    // seed gather from the just-written ent block
    gather_rows<<<blkGath, TPB, 0, stream>>>(ent_out[g], seeds[g], seed_out[g], nSeeds);
  }
}